// APPNPNet_86930138071451
// MI455X (gfx1250) — compile-verified
//
#include <hip/hip_runtime.h>
#include <hip/hip_bf16.h>

typedef __attribute__((ext_vector_type(2))) float v2f;
typedef __attribute__((ext_vector_type(8))) float v8f;
typedef __attribute__((ext_vector_type(4))) int   v4i;

typedef __attribute__((address_space(1))) v4i* gas_v4i_ptr;   // global AS
typedef __attribute__((address_space(3))) v4i* las_v4i_ptr;   // LDS AS

#define NNODES 100000
#define NEDGES 3200000
#define DIN 512
#define DHID 256
#define DOUT 64
#define KPROP 10
#define ALPHA 0.1f

// ---------------------------------------------------------------------------
// Setup kernels: degree, d^{-1/2}, edge norms (with (1-alpha) folded in)
// ---------------------------------------------------------------------------
__global__ void k_zero(float* __restrict__ p, int n) {
    int i = blockIdx.x * blockDim.x + threadIdx.x;
    if (i < n) p[i] = 0.0f;
}

__global__ void k_deg(const long long* __restrict__ dstI, float* __restrict__ deg, int E) {
    int e = blockIdx.x * blockDim.x + threadIdx.x;
    if (e < E) atomicAdd(&deg[(int)dstI[e]], 1.0f);
}

__global__ void k_dinv(const float* __restrict__ deg, float* __restrict__ dinv,
                       float* __restrict__ selfw, int n) {
    int i = blockIdx.x * blockDim.x + threadIdx.x;
    if (i < n) {
        float d = deg[i] + 1.0f;           // +1 for self loop
        float r = rsqrtf(d);               // d >= 1 always
        dinv[i] = r;
        selfw[i] = (1.0f - ALPHA) * r * r; // self-loop weight, (1-alpha) folded
    }
}

__global__ void k_norm(const long long* __restrict__ srcI, const long long* __restrict__ dstI,
                       const float* __restrict__ dinv, float* __restrict__ norm, int E) {
    int e = blockIdx.x * blockDim.x + threadIdx.x;
    if (e < E) norm[e] = (1.0f - ALPHA) * dinv[(int)srcI[e]] * dinv[(int)dstI[e]];
}

// ---------------------------------------------------------------------------
// Pack weights so each lane's WMMA B-fragment (2 consecutive K rows at one
// column) is a contiguous 8-byte pair -> single global_load_b64 per fragment.
//   PW[((k>>2)*cols + c)*4 + (k&3)] = W[k*cols + c]
// ---------------------------------------------------------------------------
__global__ void k_pack(const float* __restrict__ W, float* __restrict__ PW,
                       int cols, int total) {
    int i = blockIdx.x * blockDim.x + threadIdx.x;
    if (i >= total) return;
    int k = i / cols;
    int c = i - k * cols;
    PW[((k >> 2) * cols + c) * 4 + (k & 3)] = W[i];
}

// ---------------------------------------------------------------------------
// Fused MLP encoder: h = relu(x@W1+b1)@W2+b2 using V_WMMA_F32_16X16X4_F32.
// 4 waves / block; block handles one 16-row tile of x.
//   Stage 1: wave w -> H1 columns [w*64, w*64+64)   (4 accumulator tiles)
//   Stage 2: wave w -> OUT columns [w*16, w*16+16)  (1 accumulator tile)
// ---------------------------------------------------------------------------
__global__ __launch_bounds__(128) void k_mlp(
    const float* __restrict__ x, const float* __restrict__ pw1, const float* __restrict__ b1,
    const float* __restrict__ pw2, const float* __restrict__ b2,
    float* __restrict__ h, float* __restrict__ z0)
{
    __shared__ float ldsX[16 * DIN];    // 32 KB: x tile
    __shared__ float ldsH1[16 * DHID];  // 16 KB: relu(x@W1+b1) tile

    const int row0 = blockIdx.x * 16;
    const int tid  = threadIdx.x;
    const int wave = tid >> 5;
    const int lane = tid & 31;

    // Stage x tile into LDS (tile = 16 contiguous rows => contiguous block).
    // Use the gfx1250 async global->LDS path (ASYNCcnt, no VGPR traffic).
    {
        const float* src = x + (size_t)row0 * DIN;
#if __has_builtin(__builtin_amdgcn_global_load_async_to_lds_b128)
        for (int i = tid; i < 16 * DIN / 4; i += 128) {
            __builtin_amdgcn_global_load_async_to_lds_b128(
                (gas_v4i_ptr)(src + i * 4),
                (las_v4i_ptr)(ldsX + i * 4),
                0, 0);
        }
#if __has_builtin(__builtin_amdgcn_s_wait_asynccnt)
        __builtin_amdgcn_s_wait_asynccnt(0);
#else
        asm volatile("s_wait_asynccnt 0x0" ::: "memory");
#endif
#else
        const float4* s4 = (const float4*)src;
        float4* d4 = (float4*)ldsX;
        for (int i = tid; i < 16 * DIN / 4; i += 128) d4[i] = s4[i];
#endif
    }
    __syncthreads();

    const int mrow  = lane & 15;        // A-fragment row (M)
    const int koff  = (lane >> 4) * 2;  // lanes 0-15: K+0/K+1, lanes 16-31: K+2/K+3
    const int khalf = lane >> 4;        // packed-pair index (0 or 1)
    const int ncol  = lane & 15;        // B/C/D column within 16-wide tile
    const int mbase = (lane >> 4) * 8;  // C/D rows: r (lanes<16) or r+8 (lanes>=16)

    // ---- GEMM1: [16 x 512] @ [512 x 256] ----
    v8f acc[4] = {};
    const v2f* pB1 = (const v2f*)pw1;
    for (int kk = 0; kk < DIN / 4; ++kk) {
        v2f a;
        a.x = ldsX[mrow * DIN + kk * 4 + koff];
        a.y = ldsX[mrow * DIN + kk * 4 + koff + 1];
#pragma unroll
        for (int t = 0; t < 4; ++t) {
            const int col = wave * 64 + t * 16 + ncol;
            v2f b = pB1[(kk * DHID + col) * 2 + khalf];
            acc[t] = __builtin_amdgcn_wmma_f32_16x16x4_f32(
                false, a, false, b, (short)0, acc[t], false, false);
        }
    }

    // bias + relu -> LDS H1 tile (row-major [16][256])
#pragma unroll
    for (int t = 0; t < 4; ++t) {
        const int col = wave * 64 + t * 16 + ncol;
        const float bb = b1[col];
#pragma unroll
        for (int r = 0; r < 8; ++r) {
            float v = acc[t][r] + bb;
            ldsH1[(mbase + r) * DHID + col] = v > 0.0f ? v : 0.0f;
        }
    }
    __syncthreads();

    // ---- GEMM2: [16 x 256] @ [256 x 64] ----
    v8f acc2 = {};
    const v2f* pB2 = (const v2f*)pw2;
    const int col = wave * 16 + ncol;
    for (int kk = 0; kk < DHID / 4; ++kk) {
        v2f a;
        a.x = ldsH1[mrow * DHID + kk * 4 + koff];
        a.y = ldsH1[mrow * DHID + kk * 4 + koff + 1];
        v2f b = pB2[(kk * DOUT + col) * 2 + khalf];
        acc2 = __builtin_amdgcn_wmma_f32_16x16x4_f32(
            false, a, false, b, (short)0, acc2, false, false);
    }

    const float bb = b2[col];
#pragma unroll
    for (int r = 0; r < 8; ++r) {
        float v = acc2[r] + bb;
        size_t idx = (size_t)(row0 + mbase + r) * DOUT + col;
        h[idx]  = v;   // teleport term source
        z0[idx] = v;   // z starts at h
    }
}

// ---------------------------------------------------------------------------
// Propagation step part 1: z_new = alpha*h + selfw*z_old  (self-loop folded)
// One thread per float4 (N*16 threads).
// ---------------------------------------------------------------------------
__global__ void k_init_step(const float* __restrict__ h, const float* __restrict__ selfw,
                            const float* __restrict__ zold, float* __restrict__ znew, int n4) {
    int i = blockIdx.x * blockDim.x + threadIdx.x;
    if (i >= n4) return;
    const float sw = selfw[i >> 4];   // 16 float4s per 64-wide row
    float4 hv = ((const float4*)h)[i];
    float4 zv = ((const float4*)zold)[i];
    float4 o;
    o.x = ALPHA * hv.x + sw * zv.x;
    o.y = ALPHA * hv.y + sw * zv.y;
    o.z = ALPHA * hv.z + sw * zv.z;
    o.w = ALPHA * hv.w + sw * zv.w;
    ((float4*)znew)[i] = o;
}

// ---------------------------------------------------------------------------
// Propagation step part 2: z_new[dst] += norm * z_old[src]
// 16 lanes per edge, float4 gather + 4 f32 atomics per lane.
// ---------------------------------------------------------------------------
__global__ void k_scatter(const long long* __restrict__ srcI, const long long* __restrict__ dstI,
                          const float* __restrict__ norm, const float* __restrict__ zold,
                          float* __restrict__ znew, int E) {
    long long g = (long long)blockIdx.x * blockDim.x + threadIdx.x;
    int e = (int)(g >> 4);
    if (e >= E) return;
    const int f4 = (int)(g & 15);
    const int s = (int)srcI[e];
    const int d = (int)dstI[e];
    const float w = norm[e];
    const float4 v = ((const float4*)(zold + (size_t)s * DOUT))[f4];
    float* out = znew + (size_t)d * DOUT + f4 * 4;
    atomicAdd(out + 0, w * v.x);
    atomicAdd(out + 1, w * v.y);
    atomicAdd(out + 2, w * v.z);
    atomicAdd(out + 3, w * v.w);
}

// ---------------------------------------------------------------------------
// log_softmax over 64 columns; one wave (32 lanes) per row, 2 elems per lane
// ---------------------------------------------------------------------------
__global__ __launch_bounds__(256) void k_lsm(const float* __restrict__ z,
                                             float* __restrict__ out, int n) {
    int row = blockIdx.x * 8 + (threadIdx.x >> 5);
    if (row >= n) return;
    const int lane = threadIdx.x & 31;
    const float* p = z + (size_t)row * DOUT;
    float a = p[lane], b = p[lane + 32];
    float m = fmaxf(a, b);
#pragma unroll
    for (int off = 16; off > 0; off >>= 1) m = fmaxf(m, __shfl_xor(m, off, 32));
    float s = expf(a - m) + expf(b - m);
#pragma unroll
    for (int off = 16; off > 0; off >>= 1) s += __shfl_xor(s, off, 32);
    const float lse = m + logf(s);
    float* q = out + (size_t)row * DOUT;
    q[lane] = a - lse;
    q[lane + 32] = b - lse;
}

// ---------------------------------------------------------------------------
extern "C" void kernel_launch(void* const* d_in, const int* in_sizes, int n_in,
                              void* d_out, int out_size, void* d_ws, size_t ws_size,
                              hipStream_t stream) {
    const float* x  = (const float*)d_in[0];
    const float* W1 = (const float*)d_in[1];
    const float* b1 = (const float*)d_in[2];
    const float* W2 = (const float*)d_in[3];
    const float* b2 = (const float*)d_in[4];
    const long long* ei = (const long long*)d_in[5];   // int64 [2, E]
    const long long* srcI = ei;
    const long long* dstI = ei + NEDGES;

    float* ws    = (float*)d_ws;
    float* deg   = ws;                      // N
    float* dinv  = deg + NNODES;            // N
    float* selfw = dinv + NNODES;           // N
    float* norm  = selfw + NNODES;          // E
    float* h     = norm + NEDGES;           // N*64
    float* z0    = h + (size_t)NNODES * DOUT;
    float* z1    = z0 + (size_t)NNODES * DOUT;
    float* pw1   = z1 + (size_t)NNODES * DOUT;   // IN*HID packed W1
    float* pw2   = pw1 + (size_t)DIN * DHID;     // HID*OUT packed W2
    // total ws: ~92 MB

    const int T = 256;

    // GCN normalization
    k_zero<<<(NNODES + T - 1) / T, T, 0, stream>>>(deg, NNODES);
    k_deg<<<(NEDGES + T - 1) / T, T, 0, stream>>>(dstI, deg, NEDGES);
    k_dinv<<<(NNODES + T - 1) / T, T, 0, stream>>>(deg, dinv, selfw, NNODES);
    k_norm<<<(NEDGES + T - 1) / T, T, 0, stream>>>(srcI, dstI, dinv, norm, NEDGES);

    // Pack weights into WMMA-fragment-contiguous layout
    k_pack<<<(DIN * DHID + T - 1) / T, T, 0, stream>>>(W1, pw1, DHID, DIN * DHID);
    k_pack<<<(DHID * DOUT + T - 1) / T, T, 0, stream>>>(W2, pw2, DOUT, DHID * DOUT);

    // Fused MLP encoder (WMMA), writes h and z0
    k_mlp<<<NNODES / 16, 128, 0, stream>>>(x, pw1, b1, pw2, b2, h, z0);

    // K propagation steps, ping-pong z buffers
    float* zo = z0;
    float* zn = z1;
    const int n4 = NNODES * 16;
    const long long scatter_threads = (long long)NEDGES * 16;
    const int scatter_blocks = (int)((scatter_threads + T - 1) / T);
    for (int k = 0; k < KPROP; ++k) {
        k_init_step<<<(n4 + T - 1) / T, T, 0, stream>>>(h, selfw, zo, zn, n4);
        k_scatter<<<scatter_blocks, T, 0, stream>>>(srcI, dstI, norm, zo, zn, NEDGES);
        float* tmp = zo; zo = zn; zn = tmp;
    }

    // log_softmax -> d_out
    k_lsm<<<(NNODES + 7) / 8, T, 0, stream>>>(zo, (float*)d_out, NNODES);
}